// FBGCN_Layer_22316650070954
// MI455X (gfx1250) — compile-verified
//
#include <hip/hip_runtime.h>

#define N_NODES 12288
#define N_EDGES 196608
#define FDIM 64

typedef float v2f __attribute__((ext_vector_type(2)));
typedef float v8f __attribute__((ext_vector_type(8)));
typedef int   v4i __attribute__((vector_size(4 * sizeof(int))));

#if defined(__has_builtin)
#if __has_builtin(__builtin_amdgcn_global_load_async_to_lds_b128)
#define HAVE_ASYNC_LDS 1
#endif
#if __has_builtin(__builtin_amdgcn_s_wait_asynccnt)
#define HAVE_WAIT_ASYNC 1
#endif
#endif

// 16-byte global -> LDS copy: async (ASYNCcnt-tracked, no VGPR round trip) if
// the gfx1250 builtin exists, else synchronous float4 staging.
__device__ __forceinline__ void copy16_g2l(const float* __restrict__ g,
                                           float* __restrict__ l) {
#ifdef HAVE_ASYNC_LDS
    __builtin_amdgcn_global_load_async_to_lds_b128(
        (__attribute__((address_space(1))) v4i*)g,
        (__attribute__((address_space(3))) v4i*)l, 0, 0);
#else
    *(float4*)l = *(const float4*)g;
#endif
}

__device__ __forceinline__ void wait_async_le6() {
#ifdef HAVE_ASYNC_LDS
#ifdef HAVE_WAIT_ASYNC
    __builtin_amdgcn_s_wait_asynccnt(6);
#else
    asm volatile("s_wait_asynccnt 0x6" ::: "memory");
#endif
#endif
}

__device__ __forceinline__ void wait_async_0() {
#ifdef HAVE_ASYNC_LDS
#ifdef HAVE_WAIT_ASYNC
    __builtin_amdgcn_s_wait_asynccnt(0);
#else
    asm volatile("s_wait_asynccnt 0x0" ::: "memory");
#endif
#endif
}

// ---------------------------------------------------------------------------
// K0: init Hl accumulator to b_conv, deg to 1.0 (self loop)
// ---------------------------------------------------------------------------
__global__ void k_init(const float* __restrict__ b_conv,
                       float* __restrict__ hl, float* __restrict__ deg) {
    int gid = blockIdx.x * blockDim.x + threadIdx.x;   // N*64 threads
    int c = gid & (FDIM - 1);
    hl[gid] = b_conv[c];
    if (c == 0) deg[gid >> 6] = 1.0f;
}

// ---------------------------------------------------------------------------
// K1: edge-target degree accumulation
// ---------------------------------------------------------------------------
__global__ void k_degree(const int* __restrict__ ei, float* __restrict__ deg) {
    int e = blockIdx.x * blockDim.x + threadIdx.x;
    if (e < N_EDGES) atomicAdd(&deg[ei[N_EDGES + e]], 1.0f);
}

// ---------------------------------------------------------------------------
// K2: deg -> rsqrt(deg)   (deg >= 1 always because of self loops)
// ---------------------------------------------------------------------------
__global__ void k_dinv(float* __restrict__ deg) {
    int i = blockIdx.x * blockDim.x + threadIdx.x;
    if (i < N_NODES) deg[i] = rsqrtf(deg[i]);
}

// ---------------------------------------------------------------------------
// K3: xw = x @ Wc^T ; arelu = relu(x @ Wh^T)
// One wave per 16x16 output tile, V_WMMA_F32_16X16X4_F32, K=64 (16 steps).
// ---------------------------------------------------------------------------
__global__ void __launch_bounds__(256) k_xform(const float* __restrict__ x,
                                               const float* __restrict__ Wh,
                                               const float* __restrict__ Wc,
                                               float* __restrict__ arelu,
                                               float* __restrict__ xw) {
    int tid  = threadIdx.x;
    int wave = tid >> 5;
    int lane = tid & 31;
    int t    = blockIdx.x * 8 + wave;          // 768 blocks * 8 waves = 6144
    int mat  = t / 3072;                       // 0 = conv, 1 = high
    int r    = t % 3072;
    int mtile = r >> 2, ntile = r & 3;
    int m0 = mtile * 16, n0 = ntile * 16;

    int halfl = lane & 15;
    int koff  = (lane >> 4) << 1;              // 0 for lanes 0-15, 2 for 16-31

    const float* W = mat ? Wh : Wc;
    const float* xrow = x + (m0 + halfl) * FDIM;   // A-frag: row m0+halfl
    const float* wrow = W + (n0 + halfl) * FDIM;   // B-frag: B[k][n] = W[n][k]

    v8f c = {};
#pragma unroll
    for (int k0 = 0; k0 < FDIM; k0 += 4) {
        v2f a, b;
        a.x = xrow[k0 + koff];
        a.y = xrow[k0 + koff + 1];
        b.x = wrow[k0 + koff];
        b.y = wrow[k0 + koff + 1];
        c = __builtin_amdgcn_wmma_f32_16x16x4_f32(false, a, false, b,
                                                  (short)0, c, false, false);
    }

    float* dst = mat ? arelu : xw;
    int rbase = m0 + ((lane >> 4) << 3);       // C/D: VGPR j -> rows j, j+8
#pragma unroll
    for (int j = 0; j < 8; ++j) {
        float v = c[j];
        if (mat) v = fmaxf(v, 0.0f);
        dst[(rbase + j) * FDIM + n0 + halfl] = v;
    }
}

// ---------------------------------------------------------------------------
// K4: edge scatter   hl[col] += xw[row] * dinv[row]*dinv[col]
// ---------------------------------------------------------------------------
__global__ void k_scatter(const int* __restrict__ ei,
                          const float* __restrict__ xw,
                          const float* __restrict__ dinv,
                          float* __restrict__ hl) {
    int gid = blockIdx.x * blockDim.x + threadIdx.x;   // E*64 threads
    int e = gid >> 6, c = gid & (FDIM - 1);
    int r = ei[e];
    int t = ei[N_EDGES + e];
    float w = dinv[r] * dinv[t];
    atomicAdd(&hl[t * FDIM + c], xw[r * FDIM + c] * w);
}

// ---------------------------------------------------------------------------
// K5: Hh = Lsym @ arelu (HBM-bound, 604MB stream) fused with final combine.
// 8 waves/WG -> 32x64 output block. Double-buffered LDS K-chunks of 64 filled
// by GLOBAL_LOAD_ASYNC_TO_LDS_B128 (6 per thread per chunk), pipelined with
// s_wait_asynccnt: chunk i+1 in flight while 16 WMMAs consume chunk i.
// ---------------------------------------------------------------------------
__global__ void __launch_bounds__(256) k_hh_combine(
        const float* __restrict__ L,     const float* __restrict__ arelu,
        const float* __restrict__ hl,    const float* __restrict__ xw,
        const float* __restrict__ dinv,  const float* __restrict__ pAL,
        const float* __restrict__ pAH,   float* __restrict__ out) {
    // stride 68 floats = 272B: 16B-aligned rows, no 16-way bank conflicts
    __shared__ __align__(16) float ldsL[2][32][68];
    __shared__ __align__(16) float ldsA[2][64][68];

    int tid   = threadIdx.x;
    int mBase = blockIdx.x * 32;
    int wave  = tid >> 5, lane = tid & 31;
    int mt = wave >> 2, nt = wave & 3;           // 2 M-tiles x 4 N-tiles
    int halfl = lane & 15;
    int koff  = (lane >> 4) << 1;

    // this thread's staging coordinates (6 x 16B per chunk)
    auto issue_chunk = [&](int buf, int kc) {
#pragma unroll
        for (int i = 0; i < 2; ++i) {            // 32x64 Lsym tile
            int f = tid + i * 256;
            int row = f >> 4, c4 = (f & 15) << 2;
            copy16_g2l(&L[(mBase + row) * N_NODES + kc + c4],
                       &ldsL[buf][row][c4]);
            if (kc + 128 < N_NODES)              // prefetch 2 chunks ahead
                __builtin_prefetch(&L[(mBase + row) * N_NODES + kc + 128 + c4],
                                   0, 0);
        }
#pragma unroll
        for (int i = 0; i < 4; ++i) {            // 64x64 activation tile
            int f = tid + i * 256;
            int row = f >> 4, c4 = (f & 15) << 2;
            copy16_g2l(&arelu[(kc + row) * FDIM + c4], &ldsA[buf][row][c4]);
        }
    };

    const int nIter = N_NODES / 64;              // 192 K-chunks
    issue_chunk(0, 0);

    v8f c = {};
    for (int it = 0; it < nIter; ++it) {
        int cur = it & 1;
        if (it + 1 < nIter) {
            issue_chunk(cur ^ 1, (it + 1) * 64); // next chunk in flight
            wait_async_le6();                    // this wave's cur batch done
        } else {
            wait_async_0();
        }
        __syncthreads();                         // all waves' cur batch done
#pragma unroll
        for (int k0 = 0; k0 < 64; k0 += 4) {
            v2f a, b;
            a.x = ldsL[cur][mt * 16 + halfl][k0 + koff];
            a.y = ldsL[cur][mt * 16 + halfl][k0 + koff + 1];
            b.x = ldsA[cur][k0 + koff][nt * 16 + halfl];
            b.y = ldsA[cur][k0 + koff + 1][nt * 16 + halfl];
            c = __builtin_amdgcn_wmma_f32_16x16x4_f32(false, a, false, b,
                                                      (short)0, c, false, false);
        }
        __syncthreads();  // all waves done reading cur before it is re-filled
    }

    float aL = *pAL, aH = *pAH;
    int rbase = mBase + mt * 16 + ((lane >> 4) << 3);
    int gc    = nt * 16 + halfl;
#pragma unroll
    for (int j = 0; j < 8; ++j) {
        int g = rbase + j;
        float di  = dinv[g];
        float hlv = hl[g * FDIM + gc] + xw[g * FDIM + gc] * di * di; // + self loop
        __builtin_nontemporal_store(aL * hlv + aH * c[j], &out[g * FDIM + gc]);
    }
}

// ---------------------------------------------------------------------------
extern "C" void kernel_launch(void* const* d_in, const int* in_sizes, int n_in,
                              void* d_out, int out_size, void* d_ws, size_t ws_size,
                              hipStream_t stream) {
    const float* x     = (const float*)d_in[0];
    const int*   ei    = (const int*)d_in[1];
    const float* L     = (const float*)d_in[2];
    const float* Wh    = (const float*)d_in[3];
    const float* Wc    = (const float*)d_in[4];
    const float* bconv = (const float*)d_in[5];
    const float* aL    = (const float*)d_in[6];
    const float* aH    = (const float*)d_in[7];
    float* out = (float*)d_out;

    float* ws    = (float*)d_ws;
    float* xw    = ws;                              // N*64
    float* arelu = ws + (size_t)N_NODES * FDIM;     // N*64
    float* hl    = ws + (size_t)2 * N_NODES * FDIM; // N*64 (scatter accum)
    float* deg   = ws + (size_t)3 * N_NODES * FDIM; // N   (deg -> dinv)

    k_init   <<<(N_NODES * FDIM) / 256, 256, 0, stream>>>(bconv, hl, deg);
    k_degree <<<N_EDGES / 256,          256, 0, stream>>>(ei, deg);
    k_dinv   <<<N_NODES / 256,          256, 0, stream>>>(deg);
    k_xform  <<<768,                    256, 0, stream>>>(x, Wh, Wc, arelu, xw);
    k_scatter<<<(N_EDGES * FDIM) / 256, 256, 0, stream>>>(ei, xw, deg, hl);
    k_hh_combine<<<N_NODES / 32,        256, 0, stream>>>(L, arelu, hl, xw, deg,
                                                          aL, aH, out);
}